// Embed_37099927503248
// MI455X (gfx1250) — compile-verified
//
#include <hip/hip_runtime.h>

typedef __attribute__((ext_vector_type(2))) float v2f;
typedef __attribute__((ext_vector_type(8))) float v8f;

#define DN 32
#define DM 100
#define DK 500
#define DE 50

__global__ __launch_bounds__(256) void stlstm_interval_wmma(
    const int* __restrict__ traj_loc,   // (N,M)
    const float* __restrict__ poi,      // (LOC,2)
    const float* __restrict__ vec,      // (N,M)
    const int* __restrict__ traj_len,   // (N,)
    const int* __restrict__ cand,       // (N,K)
    const float* __restrict__ emb_su,   // (2,E)
    const float* __restrict__ emb_sl,   // (2,E)
    const float* __restrict__ emb_tu,   // (2,E)
    const float* __restrict__ emb_tl,   // (2,E)
    float* __restrict__ out)            // (N,M,K,E)
{
    const float DEG2RAD = 0.017453292519943295f;
    const float R2 = 2.0f * 6371.0f;
    const float fSU = 1000.0f, fSL = 0.0f, fTU = 500.0f, fTL = 0.0f;
    const float inv_s = 1.0f / (fSU - fSL);
    const float inv_t = 1.0f / (fTU - fTL);

    const int bid  = blockIdx.x;
    const int nm   = bid >> 2;           // (n,m) pair: 4 blocks per pair
    const int seg  = bid & 3;
    const int n    = nm / DM;
    const int m    = nm - n * DM;
    const int tid  = threadIdx.x;
    const int wave = tid >> 5;
    const int lane = tid & 31;
    const int hi   = lane >> 4;          // 0: lanes 0-15, 1: lanes 16-31
    const int sub  = lane & 15;

    // k-tile base; clamp last tile so it overlaps (duplicate identical writes,
    // no guards needed anywhere on the k dimension).
    int k0 = (seg * 8 + wave) * 16;
    if (k0 > DK - 16) k0 = DK - 16;

    // ---- per-(n,m) scalars ----
    const int   loc1 = traj_loc[n * DM + m];
    const float lat1 = poi[2 * loc1 + 0] * DEG2RAD;
    const float lon1 = poi[2 * loc1 + 1] * DEG2RAD;
    const float dt   = vec[n * DM + m];
    const int   msk  = (m < traj_len[n]) ? 1 : 0;

    // ---- haversine: all 32 lanes compute k = k0 + sub (halves duplicate) ----
    const int   k     = k0 + sub;
    const int   loc2  = cand[n * DK + k];
    const float lat2  = poi[2 * loc2 + 0] * DEG2RAD;
    const float lon2  = poi[2 * loc2 + 1] * DEG2RAD;
    const float sdlat = __sinf(0.5f * (lat2 - lat1));
    const float sdlon = __sinf(0.5f * (lon2 - lon1));
    float a = sdlat * sdlat + __cosf(lat1) * __cosf(lat2) * sdlon * sdlon;
    a = fminf(fmaxf(a, 0.0f), 1.0f);
    const float ds = R2 * asinf(sqrtf(a));

    // ---- A matrix (16x4 f32): col0 = 1, col1 = ds, cols 2-3 = 0 ----
    v2f A;
    A.x = hi ? 0.0f : 1.0f;
    A.y = hi ? 0.0f : ds;

    const float* esl = emb_sl + msk * DE;
    const float* esu = emb_su + msk * DE;
    const float* etl = emb_tl + msk * DE;
    const float* etu = emb_tu + msk * DE;

    const int obase = ((n * DM + m) * DK) * DE;          // < 2^27, fits int
    // Per-lane row base: row M = r + 8*hi of the C tile -> k = k0 + r + 8*hi.
    float* __restrict__ p = out + obase + (k0 + 8 * hi) * DE + sub;

#pragma unroll
    for (int t = 0; t < 3; ++t) {                        // e-tiles 0-15,16-31,32-47
        const int e = t * 16 + sub;                      // always < 48 < DE
        const float vsl = esl[e], vsu = esu[e];
        const float vtl = etl[e], vtu = etu[e];
        const float baseE = (vsl * fSU - vsu * fSL) * inv_s
                          + (vtl * (fTU - dt) + vtu * (dt - fTL)) * inv_t;
        const float coefE = (vsu - vsl) * inv_s;

        // B (4x16 f32): VGPR0 = row0 (lanes 0-15: base) / row2 (lanes 16-31: coef);
        // VGPR1 = rows 1/3 = zero.
        v2f B;
        B.x = hi ? coefE : baseE;
        B.y = 0.0f;

        v8f C = {};
        C = __builtin_amdgcn_wmma_f32_16x16x4_f32(
                false, A, false, B, (short)0, C, false, false);

        // 8 unconditional stores with immediate offsets (r*DE + t*16)*4 bytes.
#pragma unroll
        for (int r = 0; r < 8; ++r)
            p[r * DE + t * 16] = C[r];
    }

    // ---- e-tail (e = 48, 49): direct FMA, packed b64 store from lanes 0-15 ----
    {
        const float vsl0 = esl[48], vsu0 = esu[48], vtl0 = etl[48], vtu0 = etu[48];
        const float vsl1 = esl[49], vsu1 = esu[49], vtl1 = etl[49], vtu1 = etu[49];
        const float b48 = (vsl0 * fSU - vsu0 * fSL) * inv_s
                        + (vtl0 * (fTU - dt) + vtu0 * (dt - fTL)) * inv_t;
        const float c48 = (vsu0 - vsl0) * inv_s;
        const float b49 = (vsl1 * fSU - vsu1 * fSL) * inv_s
                        + (vtl1 * (fTU - dt) + vtu1 * (dt - fTL)) * inv_t;
        const float c49 = (vsu1 - vsl1) * inv_s;
        if (hi == 0) {
            v2f v;
            v.x = fmaf(c48, ds, b48);
            v.y = fmaf(c49, ds, b49);
            *(v2f*)(out + obase + (k0 + sub) * DE + 48) = v;   // 8B aligned
        }
    }
}

extern "C" void kernel_launch(void* const* d_in, const int* in_sizes, int n_in,
                              void* d_out, int out_size, void* d_ws, size_t ws_size,
                              hipStream_t stream) {
    const int*   traj_loc = (const int*)  d_in[0];
    const float* poi      = (const float*)d_in[1];
    const float* vec      = (const float*)d_in[2];
    const int*   traj_len = (const int*)  d_in[3];
    const int*   cand     = (const int*)  d_in[4];
    const float* emb_su   = (const float*)d_in[5];
    const float* emb_sl   = (const float*)d_in[6];
    const float* emb_tu   = (const float*)d_in[7];
    const float* emb_tl   = (const float*)d_in[8];
    float* out = (float*)d_out;

    dim3 grid(DN * DM * 4);   // 4 blocks (of 8 waves) per (n,m): 32 k-tiles of 16
    dim3 block(256);
    hipLaunchKernelGGL(stlstm_interval_wmma, grid, block, 0, stream,
                       traj_loc, poi, vec, traj_len, cand,
                       emb_su, emb_sl, emb_tu, emb_tl, out);
}